// H_GAT_79431125172510
// MI455X (gfx1250) — compile-verified
//
#include <hip/hip_runtime.h>
#include <hip/hip_bf16.h>

typedef float v2f __attribute__((ext_vector_type(2)));
typedef float v8f __attribute__((ext_vector_type(8)));

#define DIM 128

// ---------------------------------------------------------------------------
// Phase 1: per-edge message + scatter-sum.
//   one wave per edge, 4 channels per lane (32 lanes * 4 = 128 = DIM)
//   x / prop are 25.6MB each -> resident in MI455X's 192MB L2, so the random
//   gather + atomics hit L2; only edge_index/edge_weight stream from HBM.
// ---------------------------------------------------------------------------
__global__ __launch_bounds__(256) void hgat_scatter(
    const float* __restrict__ x,
    const float* __restrict__ emb,
    const long long* __restrict__ edge_index,   // [2, E] int64
    const long long* __restrict__ edge_weight,  // [E]    int64
    float* __restrict__ prop,                   // [N, DIM] pre-zeroed
    int nEdges)
{
    const int lane = threadIdx.x & 31;
    const int wave = threadIdx.x >> 5;
    const int wpb  = blockDim.x >> 5;
    const int c0   = lane << 2;  // 4 channels per lane

    for (int e = blockIdx.x * wpb + wave; e < nEdges; e += gridDim.x * wpb) {
        const int src = (int)edge_index[e];           // wave-uniform -> scalarized
        const int dst = (int)edge_index[nEdges + e];
        const int w   = (int)edge_weight[e];

        const float4 xv = *(const float4*)(x   + (long)src * DIM + c0);
        const float4 ev = *(const float4*)(emb + (long)w   * DIM + c0);

        float* p = prop + (long)dst * DIM + c0;
        __hip_atomic_fetch_add(p + 0, xv.x * ev.x, __ATOMIC_RELAXED, __HIP_MEMORY_SCOPE_AGENT);
        __hip_atomic_fetch_add(p + 1, xv.y * ev.y, __ATOMIC_RELAXED, __HIP_MEMORY_SCOPE_AGENT);
        __hip_atomic_fetch_add(p + 2, xv.z * ev.z, __ATOMIC_RELAXED, __HIP_MEMORY_SCOPE_AGENT);
        __hip_atomic_fetch_add(p + 3, xv.w * ev.w, __ATOMIC_RELAXED, __HIP_MEMORY_SCOPE_AGENT);
    }
}

// ---------------------------------------------------------------------------
// Phase 2: out = x @ W_l^T + prop @ W_r^T + (b_l + b_r) via V_WMMA_F32_16X16X4_F32
//
// LDS weight layout: for k-chunk kc4 (k = 4*kc4..4*kc4+3), half hi (0/1),
// column o, the pair (k = 4*kc4 + 2*hi, o) and (k = 4*kc4 + 2*hi + 1, o) are
// adjacent floats -> one ds_load_b64 per B fragment element.
//   idx(k, o) = ((( (k>>2)*2 + ((k>>1)&1) ) * DIM) + o) * 2 + (k & 1)
// ---------------------------------------------------------------------------
__device__ __forceinline__ void mm_accum_f32wmma(
    const float* __restrict__ A,      // global [N, DIM]
    const float* __restrict__ Bf,     // LDS, fragment-interleaved W^T
    int rowA, int m, int hi, v8f acc[8])
{
    #pragma unroll 4
    for (int kc4 = 0; kc4 < DIM / 4; ++kc4) {
        // A fragment: lane(m, hi) holds A[rowA, 4*kc4 + 2*hi + {0,1}]
        const v2f a = *(const v2f*)(A + (long)rowA * DIM + kc4 * 4 + 2 * hi);
        const float* bbase = Bf + (size_t)(kc4 * 2 + hi) * DIM * 2;
        #pragma unroll
        for (int nt = 0; nt < 8; ++nt) {
            const v2f b = *(const v2f*)(bbase + (nt * 16 + m) * 2);
            acc[nt] = __builtin_amdgcn_wmma_f32_16x16x4_f32(
                false, a, false, b, (short)0, acc[nt], false, false);
        }
    }
}

__global__ __launch_bounds__(256) void hgat_gemm(
    const float* __restrict__ x,
    const float* __restrict__ prop,
    const float* __restrict__ Wl,   // [DIM, DIM] row-major (out, in)
    const float* __restrict__ Wr,
    const float* __restrict__ bl,
    const float* __restrict__ br,
    float* __restrict__ out,
    int nNodes)
{
    extern __shared__ float lds[];
    float* sWl = lds;               // DIM*DIM floats, fragment-interleaved W_l^T
    float* sWr = lds + DIM * DIM;

    // Cooperative stage: sW[idx(k,o)] = W[o*DIM + k]
    for (int i = threadIdx.x; i < DIM * DIM; i += blockDim.x) {
        const int k = i >> 7;
        const int o = i & (DIM - 1);
        const int idx = ((((k >> 2) * 2 + ((k >> 1) & 1)) * DIM) + o) * 2 + (k & 1);
        sWl[idx] = Wl[o * DIM + k];
        sWr[idx] = Wr[o * DIM + k];
    }
    __syncthreads();

    const int lane    = threadIdx.x & 31;
    const int wave    = threadIdx.x >> 5;
    const int rowTile = blockIdx.x * (blockDim.x >> 5) + wave;   // 16 rows / tile
    const int numRowTiles = (nNodes + 15) >> 4;
    if (rowTile >= numRowTiles) return;   // wave-uniform: EXEC stays all-ones

    const int row0 = rowTile << 4;
    const int m    = lane & 15;
    const int hi   = lane >> 4;

    // accumulators initialized with bias (bias depends only on output column)
    v8f acc[8];
    #pragma unroll
    for (int nt = 0; nt < 8; ++nt) {
        const float b = bl[nt * 16 + m] + br[nt * 16 + m];
        acc[nt] = (v8f){b, b, b, b, b, b, b, b};
    }

    int rowA = row0 + m;
    if (rowA >= nNodes) rowA = nNodes - 1;   // clamp (no-op for 50000 % 16 == 0)

    mm_accum_f32wmma(x,    sWl, rowA, m, hi, acc);
    mm_accum_f32wmma(prop, sWr, rowA, m, hi, acc);

    // C/D layout: lane(m,hi), VGPR v -> row = row0 + v + 8*hi, col = nt*16 + m
    #pragma unroll
    for (int v = 0; v < 8; ++v) {
        const int r = row0 + v + 8 * hi;
        if (r < nNodes) {
            #pragma unroll
            for (int nt = 0; nt < 8; ++nt) {
                out[(long)r * DIM + nt * 16 + m] = acc[nt][v];
            }
        }
    }
}

// ---------------------------------------------------------------------------
// Inputs (setup_inputs order):
//   0: x [N,128] f32        1: edge_emb [10,128] f32
//   2: W_l [128,128] f32    3: b_l [128] f32
//   4: W_r [128,128] f32    5: b_r [128] f32
//   6: edge_index [2,E] i64 7: edge_weight [E] i64   8: cat_list (unused)
// ---------------------------------------------------------------------------
extern "C" void kernel_launch(void* const* d_in, const int* in_sizes, int n_in,
                              void* d_out, int out_size, void* d_ws, size_t ws_size,
                              hipStream_t stream)
{
    const float*     x   = (const float*)d_in[0];
    const float*     emb = (const float*)d_in[1];
    const float*     Wl  = (const float*)d_in[2];
    const float*     bl  = (const float*)d_in[3];
    const float*     Wr  = (const float*)d_in[4];
    const float*     br  = (const float*)d_in[5];
    const long long* ei  = (const long long*)d_in[6];
    const long long* ew  = (const long long*)d_in[7];

    const int nNodes = in_sizes[0] / DIM;   // 50000
    const int nEdges = in_sizes[7];         // 600000

    float* prop = (float*)d_ws;             // [N, DIM] scratch accumulator
    float* out  = (float*)d_out;

    // zero the scatter accumulator (graph-capture-safe memset node)
    hipMemsetAsync(prop, 0, (size_t)nNodes * DIM * sizeof(float), stream);

    // Phase 1: edge messages + scatter-sum (wave per edge)
    hipLaunchKernelGGL(hgat_scatter, dim3(2048), dim3(256), 0, stream,
                       x, emb, ei, ew, prop, nEdges);

    // Phase 2: fused dual-GEMM + bias with f32 WMMA
    const int numRowTiles   = (nNodes + 15) / 16;    // 3125
    const int tilesPerBlock = 256 / 32;              // 8 waves
    const int blocks        = (numRowTiles + tilesPerBlock - 1) / tilesPerBlock;
    hipLaunchKernelGGL(hgat_gemm, dim3(blocks), dim3(256),
                       2 * DIM * DIM * sizeof(float), stream,
                       x, prop, Wl, Wr, bl, br, out, nNodes);
}